// ScaledDotProductAttention_40166534152796
// MI455X (gfx1250) — compile-verified
//
#include <hip/hip_runtime.h>
#include <hip/hip_bf16.h>

typedef __attribute__((ext_vector_type(2))) float v2f;
typedef __attribute__((ext_vector_type(4))) float v4f;
typedef __attribute__((ext_vector_type(8))) float v8f;

// Problem constants (from reference): B=4, H=16, S=2048, D=64
#define SEQ   2048
#define DIM   64
#define NBH   64            // B*H
#define BQ    64            // queries per workgroup (16 per wave, 4 waves)
#define BK    64            // keys per LDS-staged block
#define NKB   (SEQ / BK)    // 32 key blocks
#define KSTR  68            // padded LDS row stride for K (floats)  -> conflict-free frag reads
#define VSTR  72            // padded LDS row stride for V (floats)
#define SSTR  68            // padded LDS row stride for S scratch

__global__ __launch_bounds__(128)
void relu_attn_wmma_f32(const float* __restrict__ Q,
                        const float* __restrict__ K,
                        const float* __restrict__ V,
                        float* __restrict__ ctx,
                        float* __restrict__ attn) {
    const int bh  = blockIdx.y;
    const int q0  = blockIdx.x * BQ;

    const float* Qb = Q + (size_t)bh * SEQ * DIM;
    const float* Kb = K + (size_t)bh * SEQ * DIM;
    const float* Vb = V + (size_t)bh * SEQ * DIM;
    float* ctxb  = ctx  + (size_t)bh * SEQ * DIM;
    float* attnb = attn + (size_t)bh * SEQ * SEQ;

    const int tid  = threadIdx.x;
    const int w    = tid >> 5;          // wave id 0..3
    const int lane = tid & 31;
    const int row  = lane & 15;         // M (A/C) or N (B/C) index
    const int half = lane >> 4;         // selects K-pair within x4 fragment
    const int c4   = (lane & 15) * 4;   // col for b128 LDS->global writeback

    __shared__ float ldsK[2][BK * KSTR];   // double-buffered K tile
    __shared__ float ldsV[2][BK * VSTR];   // double-buffered V tile
    __shared__ float ldsS[4][16 * SSTR];   // per-wave relu(S) scratch

    // ---- Load this wave's 16x64 Q stripe into A-fragments (registers) ----
    // A frag (16x4 f32): lane holds Q[row][d4*4 + half*2 + {0,1}]
    const int qrow = q0 + w * 16 + row;
    v2f qa[16];
#pragma unroll
    for (int d4 = 0; d4 < 16; ++d4)
        qa[d4] = *(const v2f*)(Qb + (size_t)qrow * DIM + d4 * 4 + half * 2);

    v8f acc[4] = {};   // context accumulators: 4 N-tiles covering d=0..63

    // Async memory->LDS staging of one 64x64 K tile + V tile.
    // Each of the 4 waves issues 8+8 GLOBAL_LOAD_ASYNC_TO_LDS_B128 (ASYNCcnt),
    // 32 lanes x 16B each: no VGPR round-trip, copies run behind compute.
    auto stage_async = [&](int kbn, int b) {
        const int r = tid >> 4;            // 0..7
        const int c = (tid & 15) * 4;      // 0..60
#pragma unroll
        for (int i = 0; i < 8; ++i) {
            const int rr = r + i * 8;
            const float* gk = Kb + (size_t)(kbn * BK + rr) * DIM + c;
            unsigned lk = (unsigned)(unsigned long long)&ldsK[b][rr * KSTR + c];
            asm volatile("global_load_async_to_lds_b128 %0, %1, off"
                         :: "v"(lk), "v"(gk) : "memory");
            const float* gv = Vb + (size_t)(kbn * BK + rr) * DIM + c;
            unsigned lv = (unsigned)(unsigned long long)&ldsV[b][rr * VSTR + c];
            asm volatile("global_load_async_to_lds_b128 %0, %1, off"
                         :: "v"(lv), "v"(gv) : "memory");
        }
    };

    stage_async(0, 0);                     // prologue: fill buffer 0

    for (int kb = 0; kb < NKB; ++kb) {
        const int buf = kb & 1;
        const int k0  = kb * BK;

        if (kb + 1 < NKB) {
            stage_async(kb + 1, buf ^ 1);  // kick off next block's copy
            // 16 outstanding belong to the next block; in-order completion
            // means <=16 guarantees the current buffer is resident.
            asm volatile("s_wait_asynccnt 0x10" ::: "memory");
        } else {
            asm volatile("s_wait_asynccnt 0x0" ::: "memory");
        }
        __syncthreads();                   // all waves' staging visible

        // ---- GEMM1: S16x16 tiles = Q stripe @ K_tile^T, over d in steps of 4 ----
#pragma unroll
        for (int n = 0; n < 4; ++n) {      // key sub-tile (16 keys)
            v8f s = {};
#pragma unroll
            for (int d4 = 0; d4 < 16; ++d4) {
                // B frag (4x16 f32) of K^T: lane holds K[n*16+row][d4*4 + half*2 + {0,1}]
                v2f bf = *(const v2f*)(&ldsK[buf][(n * 16 + row) * KSTR + d4 * 4 + half * 2]);
                s = __builtin_amdgcn_wmma_f32_16x16x4_f32(
                        false, qa[d4], false, bf, (short)0, s, false, false);
            }
            // scale by 1/sqrt(64), ReLU -> per-wave S scratch (C layout: M=e+half*8, N=row)
#pragma unroll
            for (int e = 0; e < 8; ++e) {
                float x = s[e] * 0.125f;
                x = x > 0.0f ? x : 0.0f;
                ldsS[w][(half * 8 + e) * SSTR + n * 16 + row] = x;
            }
        }

        // ---- Stream attn weights: row-linear readback, NT b128 stores ----
        // 8 x global_store_b128 th:NT per wave: two contiguous 256B rows each,
        // non-temporal so the 1.07GB attn stream doesn't evict K/V from L2.
#pragma unroll
        for (int i = 0; i < 8; ++i) {
            const int rowi = i * 2 + half;       // 0..15
            v4f t = *(const v4f*)(&ldsS[w][rowi * SSTR + c4]);
            __builtin_nontemporal_store(
                t, (v4f*)(attnb + (size_t)(q0 + w * 16 + rowi) * SEQ + k0 + c4));
        }

        // ---- GEMM2: acc += relu(S) @ V_tile, over 64 keys in steps of 4 ----
#pragma unroll
        for (int k4 = 0; k4 < 16; ++k4) {
            // A frag from S scratch: S[row][k4*4 + half*2 + {0,1}]
            v2f af = *(const v2f*)(&ldsS[w][row * SSTR + k4 * 4 + half * 2]);
#pragma unroll
            for (int n2 = 0; n2 < 4; ++n2) {
                // B frag from V: V[k4*4 + half*2 + v][n2*16 + row]
                v2f bf;
                bf.x = ldsV[buf][(k4 * 4 + half * 2 + 0) * VSTR + n2 * 16 + row];
                bf.y = ldsV[buf][(k4 * 4 + half * 2 + 1) * VSTR + n2 * 16 + row];
                acc[n2] = __builtin_amdgcn_wmma_f32_16x16x4_f32(
                              false, af, false, bf, (short)0, acc[n2], false, false);
            }
        }
        __syncthreads();   // all waves done reading buf before it is re-filled
    }

    // ---- Write context via LDS bounce: coalesced NT b128 stores ----
#pragma unroll
    for (int n2 = 0; n2 < 4; ++n2)
#pragma unroll
        for (int e = 0; e < 8; ++e)
            ldsS[w][(half * 8 + e) * SSTR + n2 * 16 + row] = acc[n2][e];
#pragma unroll
    for (int i = 0; i < 8; ++i) {
        const int rowi = i * 2 + half;
        v4f t = *(const v4f*)(&ldsS[w][rowi * SSTR + c4]);
        __builtin_nontemporal_store(
            t, (v4f*)(ctxb + (size_t)(q0 + w * 16 + rowi) * DIM + c4));
    }
}

extern "C" void kernel_launch(void* const* d_in, const int* in_sizes, int n_in,
                              void* d_out, int out_size, void* d_ws, size_t ws_size,
                              hipStream_t stream) {
    const float* Q = (const float*)d_in[0];
    const float* K = (const float*)d_in[1];
    const float* V = (const float*)d_in[2];
    float* out  = (float*)d_out;
    float* ctx  = out;                                   // [B,H,S,D] first
    float* attn = out + (size_t)NBH * SEQ * DIM;         // then [B,H,S,S]

    dim3 grid(SEQ / BQ, NBH);   // (32, 64)
    relu_attn_wmma_f32<<<grid, 128, 0, stream>>>(Q, K, V, ctx, attn);
}